// Neuralsymbol_19774029431195
// MI455X (gfx1250) — compile-verified
//
#include <hip/hip_runtime.h>
#include <math.h>

typedef __attribute__((ext_vector_type(2))) float v2f;
typedef __attribute__((ext_vector_type(8))) float v8f;

#define BLK      256
#define WAVES    8          // 256 threads / wave32
#define VDIM     128        // per-array feature width
#define KDIM     384        // 3 * 128
#define NROWS    16         // combined rows of [W_ff; W_guard]
#define WSTRIDE  388        // padded LDS stride (4-float pad -> bank-conflict-free b64 reads)
#define ROWSCR   17         // padded scratch row stride

__global__ __launch_bounds__(BLK) void fused_gate_kernel(
    const float* __restrict__ a, const float* __restrict__ b, const float* __restrict__ c,
    const float* __restrict__ Wff, const float* __restrict__ Wg,
    const float* __restrict__ W1, const float* __restrict__ b1v,
    const float* __restrict__ W2, const float* __restrict__ b2v,
    float* __restrict__ out, int nTiles)
{
    __shared__ float Wlds[NROWS * WSTRIDE];           // 16 x 384 weights, padded
    __shared__ float scr[WAVES][NROWS * ROWSCR];      // per-wave 16x16 D-tile scratch

    const int tid = threadIdx.x;

    // ---- Stage combined weight matrix Wc = [W_ff ; W_guard] into LDS ----
    for (int i = tid; i < NROWS * KDIM; i += BLK) {
        int n = i / KDIM;
        int k = i - n * KDIM;
        float w = (n < 8) ? Wff[n * KDIM + k] : Wg[(n - 8) * KDIM + k];
        Wlds[n * WSTRIDE + k] = w;
    }
    __syncthreads();

    const int lane = tid & 31;
    const int wave = tid >> 5;
    const int tile = blockIdx.x * WAVES + wave;
    if (tile >= nTiles) return;

    // A-fragment addressing (16x4 f32 WMMA layout):
    //   lanes 0-15 : row = lane,    K = {k, k+1}
    //   lanes 16-31: row = lane-16, K = {k+2, k+3}
    const int mrow = lane & 15;
    const int kh   = (lane >> 4) << 1;                // 0 or 2
    const size_t grow = (size_t)tile * 16 + mrow;

    const float* __restrict__ arow = a + grow * VDIM;
    const float* __restrict__ brow = b + grow * VDIM;
    const float* __restrict__ crow = c + grow * VDIM;

    // B-fragment addressing (4x16 f32): lane n = lane&15 holds column n,
    // VGPR0/1 = Wc[n][k+kh], Wc[n][k+kh+1]
    const int wbase = mrow * WSTRIDE + kh;

    v8f acc0 = {0.f,0.f,0.f,0.f,0.f,0.f,0.f,0.f};
    v8f acc1 = {0.f,0.f,0.f,0.f,0.f,0.f,0.f,0.f};

    const float* __restrict__ segs[3] = { arow, brow, crow };
    #pragma unroll
    for (int s = 0; s < 3; ++s) {
        const float* __restrict__ p = segs[s];
        const int wseg = wbase + s * VDIM;
        #pragma unroll
        for (int k4 = 0; k4 < VDIM; k4 += 8) {
            // step 0
            v2f af0 = *(const v2f*)(p + k4 + kh);
            v2f bf0 = { Wlds[wseg + k4],     Wlds[wseg + k4 + 1] };
            acc0 = __builtin_amdgcn_wmma_f32_16x16x4_f32(
                       false, af0, false, bf0, (short)0, acc0, false, false);
            // step 1
            v2f af1 = *(const v2f*)(p + k4 + 4 + kh);
            v2f bf1 = { Wlds[wseg + k4 + 4], Wlds[wseg + k4 + 5] };
            acc1 = __builtin_amdgcn_wmma_f32_16x16x4_f32(
                       false, af1, false, bf1, (short)0, acc1, false, false);
        }
    }

    // ---- Spill D tile to per-wave LDS scratch (transpose to row-per-lane) ----
    // D layout: VGPR r, lanes 0-15 -> (M=r,   N=lane); lanes 16-31 -> (M=r+8, N=lane-16)
    float* sc = scr[wave];
    const int rowoff = (lane >> 4) << 3;              // 0 or 8
    #pragma unroll
    for (int r = 0; r < 8; ++r)
        sc[(r + rowoff) * ROWSCR + mrow] = acc0[r] + acc1[r];

    asm volatile("s_wait_dscnt 0" ::: "memory");

    // ---- Epilogue: lanes 0-15 each own one output row ----
    if (lane < 16) {
        const float* vals = sc + lane * ROWSCR;
        float outs[8];
        #pragma unroll
        for (int j = 0; j < 8; ++j) {
            float fj = vals[j];                       // x @ W_ff.T  column j
            float gj = vals[j + 8];                   // x @ W_guard.T column j
            float sf = 1.0f / (1.0f + __expf(-fj));
            float sg = 1.0f / (1.0f + __expf(-gj));
            outs[j] = sf * sg;
        }
        // h = relu(outer(c0,c1) @ W1.T + b1), folding outer product into contraction
        float h[4];
        #pragma unroll
        for (int r = 0; r < 4; ++r) {
            float hacc = b1v[r];
            #pragma unroll
            for (int ai = 0; ai < 4; ++ai) {
                #pragma unroll
                for (int bi = 0; bi < 4; ++bi)
                    hacc = fmaf(W1[r * 16 + ai * 4 + bi], outs[ai] * outs[4 + bi], hacc);
            }
            h[r] = fmaxf(hacc, 0.0f);
        }
        float p0 = b2v[0], p1 = b2v[1];
        #pragma unroll
        for (int r = 0; r < 4; ++r) {
            p0 = fmaf(W2[r],     h[r], p0);
            p1 = fmaf(W2[4 + r], h[r], p1);
        }
        p0 = 1.0f / (1.0f + __expf(-p0));
        p1 = 1.0f / (1.0f + __expf(-p1));
        out[(size_t)tile * 16 + lane] = (p0 - p1 + 1.0f) * 0.5f;
    }
}

extern "C" void kernel_launch(void* const* d_in, const int* in_sizes, int n_in,
                              void* d_out, int out_size, void* d_ws, size_t ws_size,
                              hipStream_t stream) {
    const float* a   = (const float*)d_in[0];
    const float* b   = (const float*)d_in[1];
    const float* c   = (const float*)d_in[2];
    const float* Wff = (const float*)d_in[3];
    const float* Wg  = (const float*)d_in[4];
    const float* W1  = (const float*)d_in[5];
    const float* b1  = (const float*)d_in[6];
    const float* W2  = (const float*)d_in[7];
    const float* b2  = (const float*)d_in[8];
    float* out = (float*)d_out;

    const int B      = in_sizes[0] / VDIM;            // 524288
    const int nTiles = B / 16;                        // 32768 (B divisible by 16)
    const int blocks = (nTiles + WAVES - 1) / WAVES;  // 4096

    fused_gate_kernel<<<blocks, BLK, 0, stream>>>(a, b, c, Wff, Wg, W1, b1, W2, b2,
                                                  out, nTiles);
}